// SelfAttentionBlock_4733053960257
// MI455X (gfx1250) — compile-verified
//
#include <hip/hip_runtime.h>
#include <hip/hip_bf16.h>
#include <math.h>

// Problem constants (match reference): DIM=64, H=4, D=8, DH=32, qkv cols = 128
#define DIMC   64
#define NHEAD  4
#define QKD    8
#define DHC    32
#define QKVC   128
#define QK_SCALE 0.25f   // (DIM/H)^-0.5 = 16^-0.5

typedef __attribute__((ext_vector_type(2))) float v2f;
typedef __attribute__((ext_vector_type(8))) float v8f;

// Order-preserving float <-> uint transform so segment-max can use
// global_atomic_max_u32 (no CAS loop, correct for negatives).
__device__ __forceinline__ unsigned ford(float f) {
  unsigned u = __float_as_uint(f);
  return (u & 0x80000000u) ? ~u : (u | 0x80000000u);
}
__device__ __forceinline__ float funord(unsigned v) {
  return (v & 0x80000000u) ? __uint_as_float(v & 0x7fffffffu)
                           : __uint_as_float(~v);
}
#define NEG_INF_ORD 0x007FFFFFu  // ford(-inf)

// ---------------------------------------------------------------------------
// 0) init: zero out/deg/denom, set segment-max to -inf (ordered encoding)
// ---------------------------------------------------------------------------
__global__ __launch_bounds__(256) void init_kernel(float* __restrict__ deg,
                                                   unsigned* __restrict__ maxv,
                                                   float* __restrict__ denom,
                                                   float* __restrict__ out, int N) {
  long long i = (long long)blockIdx.x * blockDim.x + threadIdx.x;
  if (i < (long long)N * DIMC) out[i] = 0.0f;
  if (i < N) deg[i] = 0.0f;
  if (i < (long long)N * NHEAD) {
    maxv[i] = NEG_INF_ORD;
    denom[i] = 0.0f;
  }
}

// ---------------------------------------------------------------------------
// 1) qkv = x @ W + b via V_WMMA_F32_16X16X4_F32.
//    One block per 16-row tile; 8 waves cover the 8 column tiles (128 cols).
//    A-frag: lane&15 = M row, lane>=16 selects K+2;  B-frag mirrored on N.
// ---------------------------------------------------------------------------
__global__ __launch_bounds__(256) void qkv_gemm_wmma(const float* __restrict__ x,
                                                     const float* __restrict__ W,
                                                     const float* __restrict__ bias,
                                                     float* __restrict__ qkv, int N) {
  const int lane  = threadIdx.x & 31;
  const int wave  = threadIdx.x >> 5;
  const int row0  = blockIdx.x * 16;
  const int col   = wave * 16 + (lane & 15);   // output column (B/C frag N index)
  const int khalf = (lane >> 4) << 1;          // 0 or 2

  int arow = row0 + (lane & 15);
  if (arow >= N) arow = N - 1;                 // clamp loads; stores are guarded
  const float* __restrict__ ap = x + (size_t)arow * DIMC;

  v8f c = {};
#pragma unroll
  for (int kb = 0; kb < DIMC; kb += 4) {
    const int k = kb + khalf;
    v2f a; a.x = ap[k];                     a.y = ap[k + 1];
    v2f b; b.x = W[(size_t)k * QKVC + col]; b.y = W[(size_t)(k + 1) * QKVC + col];
    c = __builtin_amdgcn_wmma_f32_16x16x4_f32(false, a, false, b, (short)0, c,
                                              false, false);
  }

  const float bv    = bias[col];
  const int   mbase = (lane >> 4) << 3;        // 0 or 8
#pragma unroll
  for (int r = 0; r < 8; ++r) {
    const int row = row0 + mbase + r;
    if (row < N) qkv[(size_t)row * QKVC + col] = c[r] + bv;
  }
}

// ---------------------------------------------------------------------------
// 2) out-degree of source nodes (bincount of s)
// ---------------------------------------------------------------------------
__global__ __launch_bounds__(256) void deg_kernel(const long long* __restrict__ ei,
                                                  float* __restrict__ deg, int E) {
  int e = blockIdx.x * blockDim.x + threadIdx.x;
  if (e >= E) return;
  atomicAdd(&deg[(int)ei[e]], 1.0f);
}

// ---------------------------------------------------------------------------
// 3) compat[e,h] = <q[s]*scale, k[t]> ; segment max over s via u32 atomic max
// ---------------------------------------------------------------------------
__global__ __launch_bounds__(256) void compat_kernel(const long long* __restrict__ ei,
                                                     const float* __restrict__ qkv,
                                                     const float* __restrict__ deg,
                                                     float* __restrict__ compat,
                                                     unsigned* __restrict__ maxv,
                                                     int E) {
  int e = blockIdx.x * blockDim.x + threadIdx.x;
  if (e >= E) return;
  const int s = (int)ei[e];
  const int t = (int)ei[(size_t)E + e];
  const float scale = QK_SCALE * rsqrtf(deg[s]);
  const float* __restrict__ q = qkv + (size_t)s * QKVC;
  const float* __restrict__ k = qkv + (size_t)t * QKVC + DHC;
#pragma unroll
  for (int h = 0; h < NHEAD; ++h) {
    float acc = 0.0f;
#pragma unroll
    for (int d = 0; d < QKD; ++d) acc += q[h * QKD + d] * k[h * QKD + d];
    acc *= scale;
    compat[(size_t)e * NHEAD + h] = acc;
    atomicMax(&maxv[(size_t)s * NHEAD + h], ford(acc));
  }
}

// ---------------------------------------------------------------------------
// 4) ex = exp(compat - m[s]) in place; denom[s,h] += ex
// ---------------------------------------------------------------------------
__global__ __launch_bounds__(256) void exp_kernel(const long long* __restrict__ ei,
                                                  float* __restrict__ compat,
                                                  const unsigned* __restrict__ maxv,
                                                  float* __restrict__ denom, int E) {
  int e = blockIdx.x * blockDim.x + threadIdx.x;
  if (e >= E) return;
  const int s = (int)ei[e];
#pragma unroll
  for (int h = 0; h < NHEAD; ++h) {
    const float m  = funord(maxv[(size_t)s * NHEAD + h]);
    const float ex = __expf(compat[(size_t)e * NHEAD + h] - m);
    compat[(size_t)e * NHEAD + h] = ex;
    atomicAdd(&denom[(size_t)s * NHEAD + h], ex);
  }
}

// ---------------------------------------------------------------------------
// 5) out[s,:] += v[t,:] * attn   (one thread per (edge, component))
// ---------------------------------------------------------------------------
__global__ __launch_bounds__(256) void msg_kernel(const long long* __restrict__ ei,
                                                  const float* __restrict__ qkv,
                                                  const float* __restrict__ ex,
                                                  const float* __restrict__ denom,
                                                  float* __restrict__ out, int E) {
  long long i = (long long)blockIdx.x * blockDim.x + threadIdx.x;
  if (i >= (long long)E * DIMC) return;
  const int e = (int)(i >> 6);
  const int c = (int)(i & (DIMC - 1));
  const int h = c >> 4;  // DIM/H = 16 components per head
  const int s = (int)ei[e];
  const int t = (int)ei[(size_t)E + e];
  const float attn = ex[(size_t)e * NHEAD + h] /
                     (denom[(size_t)s * NHEAD + h] + 1e-16f);
  const float vv = qkv[(size_t)t * QKVC + 2 * DHC + c];
  atomicAdd(&out[(size_t)s * DIMC + c], vv * attn);
}

// ---------------------------------------------------------------------------
extern "C" void kernel_launch(void* const* d_in, const int* in_sizes, int n_in,
                              void* d_out, int out_size, void* d_ws, size_t ws_size,
                              hipStream_t stream) {
  const float*     x    = (const float*)d_in[0];
  const long long* ei   = (const long long*)d_in[1];  // (2,E) int64
  const float*     W    = (const float*)d_in[2];
  const float*     bvec = (const float*)d_in[3];
  float*           out  = (float*)d_out;

  const int N = in_sizes[0] / DIMC;
  const int E = in_sizes[1] / 2;

  // Workspace layout (floats): qkv[N*128] | deg[N] | maxv[N*4](u32) |
  //                            denom[N*4] | compat/ex[E*4]   (~80 MB total)
  float*    ws     = (float*)d_ws;
  float*    qkv    = ws;
  float*    deg    = qkv + (size_t)N * QKVC;
  unsigned* maxv   = (unsigned*)(deg + N);
  float*    denom  = (float*)(maxv + (size_t)N * NHEAD);
  float*    compat = denom + (size_t)N * NHEAD;

  const long long initN = (long long)N * DIMC;
  init_kernel<<<(unsigned)((initN + 255) / 256), 256, 0, stream>>>(deg, maxv,
                                                                   denom, out, N);

  const int rowTiles = (N + 15) / 16;
  qkv_gemm_wmma<<<rowTiles, 256, 0, stream>>>(x, W, bvec, qkv, N);

  const int eb = (E + 255) / 256;
  deg_kernel<<<eb, 256, 0, stream>>>(ei, deg, E);
  compat_kernel<<<eb, 256, 0, stream>>>(ei, qkv, deg, compat, maxv, E);
  exp_kernel<<<eb, 256, 0, stream>>>(ei, compat, maxv, denom, E);

  const long long msgN = (long long)E * DIMC;
  msg_kernel<<<(unsigned)((msgN + 255) / 256), 256, 0, stream>>>(ei, qkv, compat,
                                                                 denom, out, E);
}